// EdgePredictor_30399778521307
// MI455X (gfx1250) — compile-verified
//
#include <hip/hip_runtime.h>

// f32 WMMA operand types for gfx1250 (wave32):
//  A 16x4 f32 -> 2 VGPRs/lane, B 4x16 f32 -> 2 VGPRs/lane, C/D 16x16 f32 -> 8 VGPRs/lane
typedef float v2f __attribute__((ext_vector_type(2)));
typedef float v8f __attribute__((ext_vector_type(8)));

#define D_FEAT 128   // floats per node row (reference: 128)

// One wave32 per 16-edge tile.
//   lane l: edge em = l & 15, k-half = l >> 4 (64 floats each)
//   - stream u/v half-rows with b128 loads, scalar FMA partial
//   - single v_wmma_f32_16x16x4_f32 with all-ones B reduces the 2 partials
//     of each of the 16 edges (A-layout == our lane/K assignment)
//   - D rows are replicated across N columns, so each half-wave holds all 8
//     of its scores; a cndmask tree + one predicated store writes them out.
__global__ __launch_bounds__(256) void edge_dot_sigmoid_wmma(
    const float* __restrict__ h,
    const int*   __restrict__ edges_a,
    const int*   __restrict__ edges_b,
    float*       __restrict__ out,
    int E, int tilesPerSet)
{
    const int lane = threadIdx.x & 31;
    const int wave = threadIdx.x >> 5;

    int tile = blockIdx.x * (blockDim.x >> 5) + wave;
    const int* edges = edges_a;
    int outBase = 0;
    if (tile >= tilesPerSet) {            // second edge set
        edges   = edges_b;
        outBase = E;
        tile   -= tilesPerSet;
    }
    if (tile >= tilesPerSet) return;      // wave-uniform exit (grid overshoot)

    const int e0   = tile << 4;
    const int em   = lane & 15;           // edge-in-tile == A-matrix row M
    const int half = lane >> 4;           // which 64-float half of the row
    const int e    = e0 + em;
    const int ec   = (e < E) ? e : 0;     // clamp tail (rows never stored)

    const int src = edges[ec];
    const int dst = edges[(size_t)E + ec];

    const float4* __restrict__ up =
        (const float4*)(h + (size_t)src * D_FEAT) + half * (D_FEAT / 8);
    const float4* __restrict__ vp =
        (const float4*)(h + (size_t)dst * D_FEAT) + half * (D_FEAT / 8);

    float p = 0.0f;
#pragma unroll
    for (int i = 0; i < D_FEAT / 8; ++i) {   // 16 iters: 64 floats per lane
        const float4 u = up[i];
        const float4 v = vp[i];
        p = fmaf(u.x, v.x, p);
        p = fmaf(u.y, v.y, p);
        p = fmaf(u.z, v.z, p);
        p = fmaf(u.w, v.w, p);
    }

    // A[M][K]: lanes 0-15 supply K=0,1; lanes 16-31 supply K=2,3.
    // Our partial sits in the lane's first K slot, second slot = 0.
    v2f a; a[0] = p;    a[1] = 0.0f;
    v2f b; b[0] = 1.0f; b[1] = 1.0f;      // all-ones B => row sums
    v8f c = {};
    c = __builtin_amdgcn_wmma_f32_16x16x4_f32(
            /*neg_a=*/false, a, /*neg_b=*/false, b,
            /*c_mod=*/(short)0, c, /*reuse_a=*/false, /*reuse_b=*/false);

    // D layout: VGPR r holds edge r (lanes 0-15) / edge r+8 (lanes 16-31),
    // replicated across all N columns. Select c[lane&7] with a cndmask tree;
    // lanes with (lane&8)==0 are the 16 writers of 16 contiguous scores.
    const int r = lane & 7;
    const float x01 = (r & 1) ? c[1] : c[0];
    const float x23 = (r & 1) ? c[3] : c[2];
    const float x45 = (r & 1) ? c[5] : c[4];
    const float x67 = (r & 1) ? c[7] : c[6];
    const float y0  = (r & 2) ? x23 : x01;
    const float y1  = (r & 2) ? x67 : x45;
    const float x   = (r & 4) ? y1  : y0;

    const float s = __builtin_amdgcn_rcpf(1.0f + __expf(-x));  // v_rcp_f32

    if ((lane & 8) == 0) {
        const int eo = e0 + r + (half << 3);
        if (eo < E)
            out[(size_t)outBase + eo] = s;
    }
}

extern "C" void kernel_launch(void* const* d_in, const int* in_sizes, int n_in,
                              void* d_out, int out_size, void* d_ws, size_t ws_size,
                              hipStream_t stream)
{
    const float* h  = (const float*)d_in[0];
    const int*   ea = (const int*)d_in[1];   // edges_a: [2, E] flattened
    const int*   eb = (const int*)d_in[2];   // edges_b: [2, E] flattened
    float*       out = (float*)d_out;        // [score_a (E) | score_b (E)]

    const int E           = in_sizes[1] / 2;
    const int tilesPerSet = (E + 15) >> 4;
    const int totalWaves  = 2 * tilesPerSet;

    const int threads       = 256;            // 8 waves per block
    const int wavesPerBlock = threads / 32;
    const int blocks        = (totalWaves + wavesPerBlock - 1) / wavesPerBlock;

    edge_dot_sigmoid_wmma<<<blocks, threads, 0, stream>>>(
        h, ea, eb, out, E, tilesPerSet);
}